// KipfNet_res_30210799960804
// MI455X (gfx1250) — compile-verified
//
#include <hip/hip_runtime.h>

// ---------------------------------------------------------------------------
// KipfNet (ChebConv GNN + top-k pooling) for MI455X / gfx1250, f32 pipeline.
// Dense transforms: v_wmma_f32_16x16x4_f32 (codegen-confirmed), weights
// staged zero-padded+transposed in LDS so the K-loop is branch-free
// (global_load_b64 A + ds_load_b64 B + wmma, fully unrolled per shape).
// Edge propagation: L2-resident float2 gather + global_atomic_add_f32.
// ---------------------------------------------------------------------------

namespace {

constexpr int NN   = 30000;   // nodes
constexpr int EE   = 960000;  // edges
constexpr int FIN  = 16;      // input features
constexpr int H1   = 36;
constexpr int H2   = 36;
constexpr int KH   = 8;       // Chebyshev hops
constexpr int KSEL = 3000;    // ceil(0.1 * NN)
constexpr int HCAT = H1 + H2 + 2;   // 74
constexpr int NC   = 10;      // classes

typedef float v2f __attribute__((ext_vector_type(2)));
typedef float v8f __attribute__((ext_vector_type(8)));

// ---------------------------------------------------------------- utilities
__global__ void fill_f32(float* p, int n, float v) {
  int i = blockIdx.x * 256 + threadIdx.x; if (i < n) p[i] = v;
}
__global__ void fill_i32(int* p, int n, int v) {
  int i = blockIdx.x * 256 + threadIdx.x; if (i < n) p[i] = v;
}

// ------------------------------------------------------------- graph norms
__global__ void degree_main(const int* __restrict__ src, float* __restrict__ deg, int E) {
  int e = blockIdx.x * 256 + threadIdx.x;
  if (e < E) atomicAdd(&deg[src[e]], 1.0f);
}
__global__ void rsqrt_deg(const float* __restrict__ deg, float* __restrict__ dinv, int n) {
  int i = blockIdx.x * 256 + threadIdx.x;
  if (i < n) { float d = deg[i]; dinv[i] = d > 0.0f ? rsqrtf(d) : 0.0f; }
}
__global__ void compute_norm(const int* __restrict__ src, const int* __restrict__ dst,
                             const float* __restrict__ dinv, float* __restrict__ nrm, int E) {
  int e = blockIdx.x * 256 + threadIdx.x;
  if (e < E) nrm[e] = dinv[src[e]] * dinv[dst[e]];
}

// prop(h) = -sum_e norm[e] * h[src[e]] scattered to dst[e].
// One thread per (edge, feature-pair): b64 gather + 2x f32 atomics.
__global__ void prop_main(const int* __restrict__ src, const int* __restrict__ dst,
                          const float* __restrict__ nrm, const float* __restrict__ hin,
                          float* __restrict__ hout, int E, int F) {
  int F2 = F >> 1;                               // F is 16 or 36 (even)
  unsigned idx = blockIdx.x * 256u + threadIdx.x;
  unsigned tot = (unsigned)E * (unsigned)F2;
  if (idx >= tot) return;
  unsigned e  = idx / (unsigned)F2;
  unsigned f2 = idx - e * (unsigned)F2;
  int s = src[e], d = dst[e];
  float w = -nrm[e];
  v2f v = *(const v2f*)(hin + (size_t)s * F + 2u * f2);
  float* op = hout + (size_t)d * F + 2u * f2;
  atomicAdd(op,     w * v.x);
  atomicAdd(op + 1, w * v.y);
}

// t2 = 2*t2 - t0 (Chebyshev recurrence)
__global__ void cheb_combine(float* __restrict__ t2, const float* __restrict__ t0, int n) {
  int i = blockIdx.x * 256 + threadIdx.x;
  if (i < n) t2[i] = 2.0f * t2[i] - t0[i];
}

// ------------------------------------------------------- WMMA dense GEMM
// C[M x FO] (+)= A[M x FI] @ W[FI x FO] (+bias when accumulate==0).
// W staged transposed + zero-padded in LDS -> branch-free unrolled K loop.
// A overreads (pooled FI=2, final FI=74) land in zeroed workspace padding and
// multiply against LDS zeros. EXEC is all-ones at every v_wmma.
template<int FI, int FO>
__global__ __launch_bounds__(256) void gemm_wmma(
    const float* __restrict__ A, const float* __restrict__ W,
    const float* __restrict__ bias, float* __restrict__ C,
    int M, int accumulate, int ntiles, int tilesN)
{
  constexpr int KP  = (FI + 3) & ~3;             // K padded to wmma step
  constexpr int FOP = (FO + 15) & ~15;           // cols padded to tile
  __shared__ float sWt[FOP * KP];                // [col][k], zero-padded
  for (int idx = threadIdx.x; idx < FOP * KP; idx += 256) {
    int cc = idx / KP, kk = idx % KP;
    sWt[idx] = (kk < FI && cc < FO) ? W[kk * FO + cc] : 0.0f;
  }
  __syncthreads();
  int wave = blockIdx.x * 8 + (threadIdx.x >> 5);
  if (wave >= ntiles) return;                    // wave-uniform exit
  int tm = wave / tilesN, tn = wave % tilesN;
  int lane = threadIdx.x & 31;
  int half = lane >> 4, l = lane & 15;
  int row0 = tm * 16, col0 = tn * 16;
  const float* Arow = A + (size_t)(row0 + l) * FI;   // lane's A row
  const float* Wcol = sWt + (col0 + l) * KP;         // lane's B column (LDS)
  v8f acc = {};
#pragma unroll
  for (int k0 = 0; k0 < FI; k0 += 4) {
    int ka = k0 + half * 2;                      // lanes 0-15: K=k0,k0+1; 16-31: k0+2,k0+3
    v2f a = *(const v2f*)(Arow + ka);            // global_load_b64, unguarded
    v2f b = *(const v2f*)(Wcol + ka);            // ds_load_b64, zero-padded
    acc = __builtin_amdgcn_wmma_f32_16x16x4_f32(false, a, false, b,
                                                (short)0, acc, false, false);
  }
  // C/D layout: VGPR j -> row j (lanes 0-15) / row j+8 (lanes 16-31), col = lane%16
#pragma unroll
  for (int j = 0; j < 8; ++j) {
    int r = row0 + j + half * 8;
    int c = col0 + l;
    if (r < M && c < FO) {
      float base = accumulate ? C[r * FO + c] : bias[c];
      C[r * FO + c] = base + acc[j];
    }
  }
}

// -------------------------------------------------------- relu + batchnorm
__global__ void bn_stats(const float* __restrict__ X, int n, int F,
                         float* __restrict__ mu, float* __restrict__ var) {
  int c = blockIdx.x, tid = threadIdx.x;
  __shared__ float ssum[256], ssq[256];
  float s = 0.0f, q = 0.0f;
  for (int r = tid; r < n; r += 256) {
    float v = X[r * F + c]; v = v > 0.0f ? v : 0.0f;   // relu before stats
    s += v; q += v * v;
  }
  ssum[tid] = s; ssq[tid] = q; __syncthreads();
  for (int o = 128; o > 0; o >>= 1) {
    if (tid < o) { ssum[tid] += ssum[tid + o]; ssq[tid] += ssq[tid + o]; }
    __syncthreads();
  }
  if (tid == 0) {
    float m = ssum[0] / (float)n;
    mu[c] = m; var[c] = ssq[0] / (float)n - m * m;     // population variance
  }
}
__global__ void bn_apply(float* __restrict__ X, int n, int F,
                         const float* __restrict__ mu, const float* __restrict__ var,
                         const float* __restrict__ g, const float* __restrict__ be) {
  int i = blockIdx.x * 256 + threadIdx.x;
  if (i >= n * F) return;
  int c = i % F;
  float v = X[i]; v = v > 0.0f ? v : 0.0f;
  X[i] = (v - mu[c]) * rsqrtf(var[c] + 1e-5f) * g[c] + be[c];
}

// --------------------------------------------------------------- pooling
__global__ void score_kernel(const float* __restrict__ xc, const float* __restrict__ pw,
                             float* __restrict__ sc, int n) {
  int i = blockIdx.x * 256 + threadIdx.x;
  if (i >= n) return;
  float w0 = pw[0], w1 = pw[1];
  float inv = rsqrtf(w0 * w0 + w1 * w1);
  sc[i] = tanhf((xc[2 * i] * w0 + xc[2 * i + 1] * w1) * inv);
}

// Deterministic top-k: 40-step threshold bisection (resolution far below the
// order-statistic spacing) + index-ordered scan compaction.
__global__ __launch_bounds__(1024) void topk_select(const float* __restrict__ sc,
                                                    int n, int k, int* __restrict__ perm) {
  __shared__ int s_scan[1024];
  __shared__ int s_total;
  __shared__ float s_lo, s_hi;
  int tid = threadIdx.x;
  if (tid == 0) { s_lo = -1.001f; s_hi = 1.001f; }
  for (int it = 0; it < 40; ++it) {
    __syncthreads();
    float mid = 0.5f * (s_lo + s_hi);
    int cnt = 0;
    for (int i = tid; i < n; i += 1024) cnt += (sc[i] >= mid) ? 1 : 0;
    s_scan[tid] = cnt; __syncthreads();
    for (int o = 512; o > 0; o >>= 1) {
      if (tid < o) s_scan[tid] += s_scan[tid + o];
      __syncthreads();
    }
    if (tid == 0) { if (s_scan[0] >= k) s_lo = mid; else s_hi = mid; }
  }
  __syncthreads();
  float t = s_lo;                 // invariant: count(score >= t) >= k
  if (tid == 0) s_total = 0;
  __syncthreads();
  for (int base = 0; base < n; base += 1024) {
    int i = base + tid;
    int pred = (i < n && sc[i] >= t) ? 1 : 0;
    s_scan[tid] = pred;
    __syncthreads();
    for (int o = 1; o < 1024; o <<= 1) {        // Hillis-Steele inclusive scan
      int v = (tid >= o) ? s_scan[tid - o] : 0;
      __syncthreads();
      s_scan[tid] += v;
      __syncthreads();
    }
    int excl = s_scan[tid] - pred;
    int ctot = s_scan[1023];
    int pos = s_total + excl;
    if (pred && pos < k) perm[pos] = i;
    __syncthreads();
    if (tid == 0) s_total += ctot;
    __syncthreads();
  }
}

__global__ void scatter_newidx(const int* __restrict__ perm, int* __restrict__ nidx, int k) {
  int j = blockIdx.x * 256 + threadIdx.x;
  if (j < k) nidx[perm[j]] = j;
}
__global__ void xp1_kernel(const float* __restrict__ xc, const float* __restrict__ sc,
                           const int* __restrict__ perm, float* __restrict__ xp1, int k) {
  int j = blockIdx.x * 256 + threadIdx.x;
  if (j >= k) return;
  int p = perm[j]; float s = sc[p];
  xp1[2 * j]     = xc[2 * p]     * s;
  xp1[2 * j + 1] = xc[2 * p + 1] * s;
}

__global__ void degree_pool(const int* __restrict__ src, const int* __restrict__ dst,
                            const int* __restrict__ nidx, float* __restrict__ degp, int E) {
  int e = blockIdx.x * 256 + threadIdx.x;
  if (e >= E) return;
  int vs = nidx[src[e]], vd = nidx[dst[e]];
  if (vs >= 0 && vd >= 0) atomicAdd(&degp[vs], 1.0f);
}
__global__ void prop_pool(const int* __restrict__ src, const int* __restrict__ dst,
                          const int* __restrict__ nidx, const float* __restrict__ dinvp,
                          const float* __restrict__ hin, float* __restrict__ hout, int E) {
  int e = blockIdx.x * 256 + threadIdx.x;
  if (e >= E) return;
  int vs = nidx[src[e]], vd = nidx[dst[e]];
  if (vs < 0 || vd < 0) return;
  float nrm = dinvp[vs] * dinvp[vd];
  atomicAdd(&hout[2 * vd],     -nrm * hin[2 * vs]);
  atomicAdd(&hout[2 * vd + 1], -nrm * hin[2 * vs + 1]);
}

// ------------------------------------------------ nearest-neighbor unpool
__global__ __launch_bounds__(256) void nearest_kernel(const float* __restrict__ x,
    const int* __restrict__ perm, int* __restrict__ nearest, int n) {
  __shared__ float sp[2 * KSEL];                 // 24 KB pooled positions in LDS
  for (int j = threadIdx.x; j < KSEL; j += 256) {
    int p = perm[j];
    sp[2 * j]     = x[p * FIN + (FIN - 2)];
    sp[2 * j + 1] = x[p * FIN + (FIN - 1)];
  }
  __syncthreads();
  int i = blockIdx.x * 256 + threadIdx.x;
  if (i >= n) return;
  float px = x[i * FIN + (FIN - 2)], py = x[i * FIN + (FIN - 1)];
  float best = 3.4e38f; int bj = 0;
  for (int j = 0; j < KSEL; ++j) {               // first index wins ties (argmin)
    float dx = px - sp[2 * j], dy = py - sp[2 * j + 1];
    float d = dx * dx + dy * dy;
    if (d < best) { best = d; bj = j; }
  }
  nearest[i] = bj;
}

__global__ void assemble_h(const float* __restrict__ x1, const float* __restrict__ x2,
                           const float* __restrict__ xp2, const int* __restrict__ nearest,
                           float* __restrict__ h, int n) {
  int i = blockIdx.x * 256 + threadIdx.x;
  if (i >= n) return;
  float* hr = h + (size_t)i * HCAT;
  const float* a = x1 + (size_t)i * H1;
  const float* b = x2 + (size_t)i * H2;
  for (int c = 0; c < H1; ++c) hr[c] = a[c];
  for (int c = 0; c < H2; ++c) hr[H1 + c] = b[c];
  int j = nearest[i];
  hr[H1 + H2]     = xp2[2 * j];
  hr[H1 + H2 + 1] = xp2[2 * j + 1];
}

// ----------------------------------------------------------- host helpers
template<int FI, int FO>
inline void launch_gemm(hipStream_t st, const float* A, const float* Wk, const float* b,
                        float* C, int M, int acc) {
  int tm = (M + 15) / 16, tn = (FO + 15) / 16, nt = tm * tn;
  gemm_wmma<FI, FO><<<(nt + 7) / 8, 256, 0, st>>>(A, Wk, b, C, M, acc, nt, tn);
}

// Full ChebConv: OUT = sum_k T_k(Lhat) X @ W[k] + b  with T_0=X, T_1=prop(X),
// T_k = 2*prop(T_{k-1}) - T_{k-2}.  nidx==nullptr -> main graph (uses nrm),
// else pooled graph (uses dinvp via prop_pool).
template<int FI, int FO>
inline void cheb(hipStream_t st, const float* X, float* OUT, const float* Wt, const float* b,
                 int M, const int* src, const int* dst, int E,
                 const float* nrm_or_dinvp, const int* nidx,
                 float* bA, float* bB, float* bC) {
  int zn = ((M + 15) & ~15) * FI + 16;   // zero incl. tile/K padding region
  auto zero = [&](float* p) { fill_f32<<<(zn + 255) / 256, 256, 0, st>>>(p, zn, 0.0f); };
  auto prop = [&](const float* in, float* out) {
    if (nidx) {
      prop_pool<<<(E + 255) / 256, 256, 0, st>>>(src, dst, nidx, nrm_or_dinvp, in, out, E);
    } else {
      unsigned tot = (unsigned)E * (unsigned)(FI / 2);
      prop_main<<<(tot + 255u) / 256u, 256, 0, st>>>(src, dst, nrm_or_dinvp, in, out, E, FI);
    }
  };
  launch_gemm<FI, FO>(st, X, Wt, b, OUT, M, 0);          // k=0 (+bias)
  zero(bA);
  prop(X, bA);                                           // T_1
  launch_gemm<FI, FO>(st, bA, Wt + 1 * FI * FO, b, OUT, M, 1);
  const float* t0 = X; float* t1 = bA;
  float* bufs[3] = { bA, bB, bC };
  for (int kk = 2; kk < KH; ++kk) {
    float* t2 = bufs[(kk + 2) % 3];
    zero(t2);
    prop(t1, t2);
    cheb_combine<<<(M * FI + 255) / 256, 256, 0, st>>>(t2, t0, M * FI);
    launch_gemm<FI, FO>(st, t2, Wt + kk * FI * FO, b, OUT, M, 1);
    t0 = t1; t1 = t2;
  }
}

} // anonymous namespace

// ---------------------------------------------------------------------------
extern "C" void kernel_launch(void* const* d_in, const int* in_sizes, int n_in,
                              void* d_out, int out_size, void* d_ws, size_t ws_size,
                              hipStream_t stream) {
  (void)in_sizes; (void)n_in; (void)out_size; (void)ws_size;
  const float* x   = (const float*)d_in[0];
  const int*   ei  = (const int*)d_in[1];
  const float* W1  = (const float*)d_in[2];
  const float* b1  = (const float*)d_in[3];
  const float* W2  = (const float*)d_in[4];
  const float* b2  = (const float*)d_in[5];
  const float* g1  = (const float*)d_in[6];
  const float* be1 = (const float*)d_in[7];
  const float* g2  = (const float*)d_in[8];
  const float* be2 = (const float*)d_in[9];
  const float* Wc  = (const float*)d_in[10];
  const float* bc  = (const float*)d_in[11];
  const float* pw  = (const float*)d_in[12];
  const float* Wp  = (const float*)d_in[13];
  const float* bp  = (const float*)d_in[14];
  const float* W3  = (const float*)d_in[15];
  const float* b3  = (const float*)d_in[16];
  const int* src = ei;
  const int* dst = ei + EE;

  // workspace carve-out (floats/ints, 8-byte aligned; padded for overreads)
  float* base = (float*)d_ws; size_t off = 0;
  auto alloc = [&](size_t n) { float* p = base + off; off += n; return p; };
  float* deg   = alloc(NN);
  float* dinv  = alloc(NN);
  float* nrm   = alloc(EE);
  float* bA    = alloc((size_t)NN * H1 + 64);
  float* bB    = alloc((size_t)NN * H1 + 64);
  float* bC    = alloc((size_t)NN * H1 + 64);
  float* x1    = alloc((size_t)NN * H1);
  float* x2    = alloc((size_t)NN * H2);
  float* xc    = alloc((size_t)NN * 2);
  float* sc    = alloc(NN);
  int*   perm  = (int*)alloc(3008);
  int*   nidx  = (int*)alloc(NN);
  float* xp1   = alloc(3008 * 2 + 32);
  float* degp  = alloc(3008);
  float* dinvp = alloc(3008);
  float* xp2   = alloc(3008 * 2 + 32);
  int*   nrst  = (int*)alloc(NN);
  float* h     = alloc((size_t)NN * HCAT + 64);
  float* mu    = alloc(64);
  float* var   = alloc(64);

  // symmetric normalization of the full graph (valid == 1 everywhere)
  fill_f32<<<(NN + 255) / 256, 256, 0, stream>>>(deg, NN, 0.0f);
  degree_main<<<(EE + 255) / 256, 256, 0, stream>>>(src, deg, EE);
  rsqrt_deg<<<(NN + 255) / 256, 256, 0, stream>>>(deg, dinv, NN);
  compute_norm<<<(EE + 255) / 256, 256, 0, stream>>>(src, dst, dinv, nrm, EE);

  // x1 = BN(relu(cheb1(x)))
  cheb<FIN, H1>(stream, x, x1, W1, b1, NN, src, dst, EE, nrm, nullptr, bA, bB, bC);
  bn_stats<<<H1, 256, 0, stream>>>(x1, NN, H1, mu, var);
  bn_apply<<<(NN * H1 + 255) / 256, 256, 0, stream>>>(x1, NN, H1, mu, var, g1, be1);

  // x2 = BN(relu(cheb2(x1)))
  cheb<H1, H2>(stream, x1, x2, W2, b2, NN, src, dst, EE, nrm, nullptr, bA, bB, bC);
  bn_stats<<<H2, 256, 0, stream>>>(x2, NN, H2, mu, var);
  bn_apply<<<(NN * H2 + 255) / 256, 256, 0, stream>>>(x2, NN, H2, mu, var, g2, be2);

  // scoring + deterministic top-k pooling
  launch_gemm<H1, 2>(stream, x1, Wc, bc, xc, NN, 0);          // K=1 conv
  score_kernel<<<(NN + 255) / 256, 256, 0, stream>>>(xc, pw, sc, NN);
  fill_i32<<<(3008 + 255) / 256, 256, 0, stream>>>(perm, 3008, 0);
  topk_select<<<1, 1024, 0, stream>>>(sc, NN, KSEL, perm);
  fill_i32<<<(NN + 255) / 256, 256, 0, stream>>>(nidx, NN, -1);
  scatter_newidx<<<(KSEL + 255) / 256, 256, 0, stream>>>(perm, nidx, KSEL);
  fill_f32<<<(3008 * 2 + 32 + 255) / 256, 256, 0, stream>>>(xp1, 3008 * 2 + 32, 0.0f);
  xp1_kernel<<<(KSEL + 255) / 256, 256, 0, stream>>>(xc, sc, perm, xp1, KSEL);

  // pooled-graph normalization + pooled ChebConv
  fill_f32<<<(3008 + 255) / 256, 256, 0, stream>>>(degp, 3008, 0.0f);
  degree_pool<<<(EE + 255) / 256, 256, 0, stream>>>(src, dst, nidx, degp, EE);
  rsqrt_deg<<<(KSEL + 255) / 256, 256, 0, stream>>>(degp, dinvp, KSEL);
  cheb<2, 2>(stream, xp1, xp2, Wp, bp, KSEL, src, dst, EE, dinvp, nidx, bA, bB, bC);

  // unpool + concat + final K=1 conv into d_out [30000 x 10]
  nearest_kernel<<<(NN + 255) / 256, 256, 0, stream>>>(x, perm, nrst, NN);
  assemble_h<<<(NN + 255) / 256, 256, 0, stream>>>(x1, x2, xp2, nrst, h, NN);
  fill_f32<<<1, 64, 0, stream>>>(h + (size_t)NN * HCAT, 64, 0.0f);   // zero K-pad overread region
  launch_gemm<HCAT, NC>(stream, h, W3, b3, (float*)d_out, NN, 0);
}